// PostprocessingLayer_1082331759351
// MI455X (gfx1250) — compile-verified
//
#include <hip/hip_runtime.h>
#include <stdint.h>

// Problem constants (match reference)
#define BB 2
#define HH 512
#define WW 512
#define HWSZ (HH * WW)
#define KP 128          // MAX_PEAKS
#define WIN 10          // floor(PEAK_MIN_DISTANCE / spacing)
#define MINCNT 5.0f
#define MAXC 2048       // candidate cap per batch

typedef __attribute__((ext_vector_type(2))) float v2f;
typedef __attribute__((ext_vector_type(8))) float v8f;

// ---------------- 0) zero votes + candidate counters ----------------
__global__ void zero_kernel(float* votes, int* candCount) {
    int i = blockIdx.x * blockDim.x + threadIdx.x;
    if (i < BB * HWSZ) votes[i] = 0.0f;
    if (i < BB) candCount[i] = 0;
}

// ---------------- 1) vote accumulation (scatter-add) ----------------
__global__ void vote_kernel(const float2* __restrict__ emb,
                            const unsigned char* __restrict__ fg,
                            float* __restrict__ votes) {
    int i = blockIdx.x * blockDim.x + threadIdx.x;
    if (i >= BB * HWSZ) return;
    if (!fg[i]) return;
    int b = i / HWSZ;
    float2 e = emb[i];                       // e.x = y-coord, e.y = x-coord
    int y = (int)rintf(e.x); y = y < 0 ? 0 : (y > HH - 1 ? HH - 1 : y);
    int x = (int)rintf(e.y); x = x < 0 ? 0 : (x > WW - 1 ? WW - 1 : x);
    atomicAdd(&votes[b * HWSZ + y * WW + x], 1.0f);
}

// ---------------- 2a) separable max-pool: row pass ----------------
__global__ void rowmax_kernel(const float* __restrict__ votes,
                              float* __restrict__ rowmax) {
    int i = blockIdx.x * blockDim.x + threadIdx.x;
    if (i >= BB * HWSZ) return;
    int x = i % WW;
    int rowBase = i - x;
    float m = -3.0e38f;
    int x0 = x - WIN < 0 ? 0 : x - WIN;
    int x1 = x + WIN > WW - 1 ? WW - 1 : x + WIN;
    for (int xx = x0; xx <= x1; ++xx) m = fmaxf(m, votes[rowBase + xx]);
    rowmax[i] = m;
}

// ---------------- 2b) col pass + candidate compaction ----------------
__global__ void colmax_peaks_kernel(const float* __restrict__ votes,
                                    const float* __restrict__ rowmax,
                                    float* __restrict__ candVal,
                                    int* __restrict__ candIdx,
                                    int* __restrict__ candCount) {
    int i = blockIdx.x * blockDim.x + threadIdx.x;
    if (i >= BB * HWSZ) return;
    int b = i / HWSZ;
    int p = i - b * HWSZ;
    int y = p / WW, x = p % WW;
    float pooled = -3.0e38f;
    int y0 = y - WIN < 0 ? 0 : y - WIN;
    int y1 = y + WIN > HH - 1 ? HH - 1 : y + WIN;
    for (int yy = y0; yy <= y1; ++yy)
        pooled = fmaxf(pooled, rowmax[b * HWSZ + yy * WW + x]);
    float v = votes[i];
    if (v >= pooled && v >= MINCNT) {
        int pos = atomicAdd(&candCount[b], 1);
        if (pos < MAXC) {
            candVal[b * MAXC + pos] = v;
            candIdx[b * MAXC + pos] = p;
        }
    }
}

// ---------------- 3) per-batch top-128 + stage A-matrix tiles ----------------
// Staged layout matches WMMA 16x4 f32 A operand:
//   lane m    : vgpr0 = A[m,0] = -2*py , vgpr1 = A[m,1] = -2*px
//   lane m+16 : vgpr0 = A[m,2] = p2    , vgpr1 = A[m,3] = 0
// stagedA[b*512 + tile*64 + vgpr*32 + lane]
__global__ void topk_stage_kernel(float* __restrict__ candVal,
                                  const int* __restrict__ candIdx,
                                  const int* __restrict__ candCount,
                                  float* __restrict__ stagedA,
                                  int* __restrict__ hasPeak) {
    __shared__ float rv[128]; __shared__ int ri[128]; __shared__ int rp[128];
    __shared__ float selY[KP], selX[KP]; __shared__ int selValid[KP];
    int b = blockIdx.x;
    int t = threadIdx.x;
    int cnt = candCount[b]; if (cnt > MAXC) cnt = MAXC;
    float* cv = candVal + b * MAXC;
    const int* ci = candIdx + b * MAXC;

    for (int slot = 0; slot < KP; ++slot) {
        float bv = -1.0f; int bi = 0x7fffffff; int bp = -1;
        for (int j = t; j < cnt; j += 128) {
            float v = cv[j]; int id = ci[j];
            if (v > bv || (v == bv && id < bi)) { bv = v; bi = id; bp = j; }
        }
        rv[t] = bv; ri[t] = bi; rp[t] = bp;
        __syncthreads();
        for (int s = 64; s > 0; s >>= 1) {
            if (t < s) {
                if (rv[t + s] > rv[t] || (rv[t + s] == rv[t] && ri[t + s] < ri[t])) {
                    rv[t] = rv[t + s]; ri[t] = ri[t + s]; rp[t] = rp[t + s];
                }
            }
            __syncthreads();
        }
        if (t == 0) {
            if (rv[0] > 0.0f) {
                selValid[slot] = 1;
                selY[slot] = (float)(ri[0] / WW);
                selX[slot] = (float)(ri[0] % WW);
                cv[rp[0]] = -2.0f;              // mark consumed
            } else {
                selValid[slot] = 0;
            }
        }
        __syncthreads();
    }

    if (t < KP) {
        int tile = t >> 4, m = t & 15;
        int valid = selValid[t];
        float py = valid ? selY[t] : 0.0f;
        float px = valid ? selX[t] : 0.0f;
        float a0 = valid ? -2.0f * py : 0.0f;
        float a1 = valid ? -2.0f * px : 0.0f;
        float a2 = valid ? (py * py + px * px) : 1.0e30f;  // masks invalid peaks
        float* st = stagedA + b * 512 + tile * 64;
        st[0 * 32 + m]      = a0;
        st[1 * 32 + m]      = a1;
        st[0 * 32 + m + 16] = a2;
        st[1 * 32 + m + 16] = 0.0f;
    }
    if (t == 0) hasPeak[b] = selValid[0];
}

// ---------------- 4) WMMA clustering: argmin over 128 peaks ----------------
// One wave handles 16 pixels. B operand (4x16 f32):
//   lane n    : vgpr0 = ey, vgpr1 = ex
//   lane n+16 : vgpr0 = 1 , vgpr1 = 0
// D[m,n] = p2[m] - 2*(py[m]*ey + px[m]*ex); lane n holds rows 0..7 in 8 vgprs,
// lane n+16 holds rows 8..15 for the same pixel column n.
__global__ void cluster_wmma_kernel(const float2* __restrict__ emb,
                                    const unsigned char* __restrict__ fg,
                                    const float* __restrict__ stagedA,
                                    const int* __restrict__ hasPeak,
                                    int* __restrict__ out) {
    const int blocksPerBatch = HWSZ / 128;   // 8 waves * 16 pixels per block
    int lane = threadIdx.x & 31;
    int wave = threadIdx.x >> 5;
    int b = blockIdx.x / blocksPerBatch;
    int base = (blockIdx.x % blocksPerBatch) * 128 + wave * 16;
    bool hi = lane >= 16;
    int l16 = lane & 15;
    int pix = b * HWSZ + base + l16;

    float2 e = emb[pix];
    v2f bvec;
    bvec.x = hi ? 1.0f : e.x;
    bvec.y = hi ? 0.0f : e.y;

    const float* st = stagedA + b * 512;
    float bestV = 3.0e38f;
    int bestI = 0;
    v8f c = {};

#pragma unroll
    for (int t = 0; t < 8; ++t) {
        v2f a;
        a.x = st[t * 64 + lane];
        a.y = st[t * 64 + 32 + lane];
        v8f d = __builtin_amdgcn_wmma_f32_16x16x4_f32(
            /*neg_a=*/false, a, /*neg_b=*/false, bvec,
            /*c_mod=*/(short)0, c, /*reuse_a=*/false, /*reuse_b=*/false);
        int ibase = t * 16 + (hi ? 8 : 0);
#pragma unroll
        for (int r = 0; r < 8; ++r) {
            float v = d[r];
            if (v < bestV) { bestV = v; bestI = ibase + r; }  // strict < keeps first min
        }
    }

    // merge hi half (peaks m+8) into lo half (peaks m) with argmin tie-break
    float ov = __shfl_down(bestV, 16, 32);
    int   oi = __shfl_down(bestI, 16, 32);
    if (ov < bestV || (ov == bestV && oi < bestI)) { bestV = ov; bestI = oi; }

    if (!hi) {
        int label = (fg[pix] && hasPeak[b]) ? (bestI + 1) : 0;
        out[pix] = label;
    }
}

// ---------------- launch ----------------
static inline size_t alignUp(size_t x, size_t a) { return (x + a - 1) & ~(a - 1); }

extern "C" void kernel_launch(void* const* d_in, const int* in_sizes, int n_in,
                              void* d_out, int out_size, void* d_ws, size_t ws_size,
                              hipStream_t stream) {
    const float2* emb = (const float2*)d_in[0];              // [B,H,W,2] f32
    const unsigned char* fg = (const unsigned char*)d_in[1]; // [B,H,W] bool (1B)
    int* out = (int*)d_out;                                  // [B,H,W] int32 labels

    char* ws = (char*)d_ws;
    size_t off = 0;
    float* votes   = (float*)(ws + off); off = alignUp(off + sizeof(float) * BB * HWSZ, 256);
    float* rowmax  = (float*)(ws + off); off = alignUp(off + sizeof(float) * BB * HWSZ, 256);
    float* candVal = (float*)(ws + off); off = alignUp(off + sizeof(float) * BB * MAXC, 256);
    int*   candIdx = (int*)  (ws + off); off = alignUp(off + sizeof(int)   * BB * MAXC, 256);
    int*   candCnt = (int*)  (ws + off); off = alignUp(off + sizeof(int)   * BB, 256);
    float* stagedA = (float*)(ws + off); off = alignUp(off + sizeof(float) * BB * 512, 256);
    int*   hasPeak = (int*)  (ws + off); off = alignUp(off + sizeof(int)   * BB, 256);
    (void)ws_size; (void)in_sizes; (void)n_in; (void)out_size;

    const int N = BB * HWSZ;
    const int TB = 256;
    const int NB = (N + TB - 1) / TB;

    zero_kernel<<<NB, TB, 0, stream>>>(votes, candCnt);
    vote_kernel<<<NB, TB, 0, stream>>>(emb, fg, votes);
    rowmax_kernel<<<NB, TB, 0, stream>>>(votes, rowmax);
    colmax_peaks_kernel<<<NB, TB, 0, stream>>>(votes, rowmax, candVal, candIdx, candCnt);
    topk_stage_kernel<<<BB, 128, 0, stream>>>(candVal, candIdx, candCnt, stagedA, hasPeak);
    cluster_wmma_kernel<<<BB * (HWSZ / 128), 256, 0, stream>>>(emb, fg, stagedA, hasPeak, out);
}